// RecessionGATTransformer_16174846837131
// MI455X (gfx1250) — compile-verified
//
#include <hip/hip_runtime.h>
#include <cstdint>

typedef float v2f __attribute__((ext_vector_type(2)));
typedef float v8f __attribute__((ext_vector_type(8)));

#define IN_F 128
#define OUT_F 32
#define HEADS 4
#define CH (HEADS * OUT_F) /* 128 */
#define EDGE_DIM 8
#define ALPHA 0.2f

// -------- monotone float<->uint mapping so integer atomicMax == float max ----
__device__ __forceinline__ unsigned f32_ordered(float f) {
  unsigned u = __float_as_uint(f);
  return (u & 0x80000000u) ? ~u : (u | 0x80000000u);
}
__device__ __forceinline__ float ordered_f32(unsigned o) {
  return (o & 0x80000000u) ? __uint_as_float(o & 0x7FFFFFFFu)
                           : __uint_as_float(~o);
}
#define ORDERED_NEGINF 0x007FFFFFu // f32_ordered(-inf)

// -------- kernel 1: xt[n, h*32+o] = sum_k x[n,k] * W[h,k,o] -----------------
// M = n_nodes (mult of 16), N = 128, K = 128 via V_WMMA_F32_16X16X4_F32.
// Block = 256 threads = 8 waves; wave w owns N-tile w; block owns 16 M-rows.
__global__ void gat_gemm_wmma(const float* __restrict__ x,
                              const float* __restrict__ W,
                              float* __restrict__ xt) {
  const int wave = threadIdx.x >> 5;  // 0..7 -> which 16-col tile of N=128
  const int lane = threadIdx.x & 31;
  const int half = lane >> 4;         // K-pair select for A/B fragments
  const int lid  = lane & 15;
  const int mBase = blockIdx.x << 4;

  const int col = (wave << 4) + lid;  // global output column 0..127
  const int h = col >> 5;             // head
  const int o = col & 31;             // out-feature within head

  const float* aRow = x + (size_t)(mBase + lid) * IN_F;        // A: row-major x
  const float* bCol = W + (size_t)h * (IN_F * OUT_F) + o;      // B: W[h,k,o], k-stride 32

  v8f acc = {};
#pragma unroll 8
  for (int k0 = 0; k0 < IN_F; k0 += 4) {
    const int kA = k0 + (half << 1);
    // A fragment: lanes 0-15 hold (K=k0,k0+1), lanes 16-31 hold (K=k0+2,k0+3)
    v2f aF = *reinterpret_cast<const v2f*>(aRow + kA);  // 8B aligned (kA even)
    // B fragment mirrors A layout over K, striped across N in lanes
    v2f bF;
    bF.x = bCol[(size_t)kA * OUT_F];
    bF.y = bCol[(size_t)(kA + 1) * OUT_F];
    acc = __builtin_amdgcn_wmma_f32_16x16x4_f32(false, aF, false, bF,
                                                (short)0, acc, false, false);
  }

  // D layout: VGPR r -> M = r (lanes 0-15) / M = r+8 (lanes 16-31), N = lid
  float* outRow = xt + (size_t)(mBase + (half << 3)) * CH + col;
#pragma unroll
  for (int r = 0; r < 8; ++r)
    outRow[(size_t)r * CH] = acc[r];
}

// -------- kernel 2: per-node attention scalars ------------------------------
// s[n,h,0] = xt[n,h,:] . a_s[h]   s[n,h,1] = xt[n,h,:] . a_d[h]
__global__ void gat_node_scalars(const float* __restrict__ xt,
                                 const float* __restrict__ a,
                                 float* __restrict__ s, int n_nh) {
  int t = blockIdx.x * blockDim.x + threadIdx.x;
  if (t >= n_nh) return;
  int n = t >> 2, h = t & 3;
  const float* xr = xt + (size_t)n * CH + h * OUT_F;
  const float* av = a + (size_t)h * (2 * OUT_F + EDGE_DIM);
  float ss = 0.f, sd = 0.f;
#pragma unroll
  for (int o = 0; o < OUT_F; ++o) {
    float v = xr[o];
    ss += v * av[o];
    sd += v * av[OUT_F + o];
  }
  s[(size_t)n * (2 * HEADS) + h * 2 + 0] = ss;
  s[(size_t)n * (2 * HEADS) + h * 2 + 1] = sd;
}

// -------- init: zero output, -inf maxima, zero sums -------------------------
__global__ void gat_init(float* __restrict__ out, unsigned* __restrict__ maxb,
                         float* __restrict__ sumb, int n_out, int n_nh) {
  int i = blockIdx.x * blockDim.x + threadIdx.x;
  if (i < n_out) out[i] = 0.f;
  if (i < n_nh) {
    maxb[i] = ORDERED_NEGINF;
    sumb[i] = 0.f;
  }
}

// -------- kernel 3: edge logits + leaky-relu + segment max ------------------
__global__ void gat_edge_logits(const int* __restrict__ ei,
                                const float* __restrict__ ea,
                                const float* __restrict__ a,
                                const float* __restrict__ s,
                                float* __restrict__ ebuf,
                                unsigned* __restrict__ maxb, int n_edges) {
  int e = blockIdx.x * blockDim.x + threadIdx.x;
  if (e >= n_edges) return;
  int src = ei[e];
  int dst = ei[n_edges + e];
  const float4* e4 = reinterpret_cast<const float4*>(ea + (size_t)e * EDGE_DIM);
  float4 p0 = e4[0], p1 = e4[1];
  float ev[EDGE_DIM] = {p0.x, p0.y, p0.z, p0.w, p1.x, p1.y, p1.z, p1.w};
#pragma unroll
  for (int h = 0; h < HEADS; ++h) {
    float val = s[(size_t)src * (2 * HEADS) + h * 2 + 0] +
                s[(size_t)dst * (2 * HEADS) + h * 2 + 1];
    const float* ae = a + (size_t)h * (2 * OUT_F + EDGE_DIM) + 2 * OUT_F;
#pragma unroll
    for (int d = 0; d < EDGE_DIM; ++d) val += ev[d] * ae[d];
    val = val > 0.f ? val : ALPHA * val;  // leaky relu
    ebuf[(size_t)e * HEADS + h] = val;
    atomicMax(&maxb[(size_t)dst * HEADS + h], f32_ordered(val));
  }
}

// -------- kernel 4: decode ordered max, -inf -> 0 (in place as float) -------
__global__ void gat_max_fix(unsigned* __restrict__ maxb, int n_nh) {
  int i = blockIdx.x * blockDim.x + threadIdx.x;
  if (i >= n_nh) return;
  float f = ordered_f32(maxb[i]);
  if (f == -__builtin_inff()) f = 0.f;
  reinterpret_cast<float*>(maxb)[i] = f;
}

// -------- kernel 5: exp(e - max[dst]) + segment sum -------------------------
__global__ void gat_edge_exp(const int* __restrict__ ei,
                             const float* __restrict__ maxf,
                             float* __restrict__ ebuf,
                             float* __restrict__ sumb, int n_edges) {
  int e = blockIdx.x * blockDim.x + threadIdx.x;
  if (e >= n_edges) return;
  int dst = ei[n_edges + e];
#pragma unroll
  for (int h = 0; h < HEADS; ++h) {
    float ex = __expf(ebuf[(size_t)e * HEADS + h] -
                      maxf[(size_t)dst * HEADS + h]);
    ebuf[(size_t)e * HEADS + h] = ex;
    atomicAdd(&sumb[(size_t)dst * HEADS + h], ex);
  }
}

// -------- kernel 6: out[dst] += att * xt[src], one wave32 per edge ----------
// lane = channel within head per iteration -> coalesced 128B load + atomic row
__global__ void gat_aggregate(const int* __restrict__ ei,
                              const float* __restrict__ xt,
                              const float* __restrict__ ebuf,
                              const float* __restrict__ sumb,
                              float* __restrict__ out, int n_edges) {
  int gw = (int)((blockIdx.x * blockDim.x + threadIdx.x) >> 5);
  int lane = threadIdx.x & 31;
  if (gw >= n_edges) return;
  int src = ei[gw];
  int dst = ei[n_edges + gw];
  const float* xr = xt + (size_t)src * CH;
  float* orow = out + (size_t)dst * CH;
#pragma unroll
  for (int h = 0; h < HEADS; ++h) {
    float att = ebuf[(size_t)gw * HEADS + h] /
                (sumb[(size_t)dst * HEADS + h] + 1e-10f);
    int c = h * OUT_F + lane;  // OUT_F==32 -> one head per wave iteration
    atomicAdd(&orow[c], att * xr[c]);
  }
}

extern "C" void kernel_launch(void* const* d_in, const int* in_sizes, int n_in,
                              void* d_out, int out_size, void* d_ws,
                              size_t ws_size, hipStream_t stream) {
  const float* x  = (const float*)d_in[0];
  const int*   ei = (const int*)d_in[1];
  const float* ea = (const float*)d_in[2];
  const float* W  = (const float*)d_in[3];
  const float* a  = (const float*)d_in[4];
  float* out = (float*)d_out;

  const int n_nodes = in_sizes[0] / IN_F;  // 50000 (multiple of 16)
  const int n_edges = in_sizes[1] / 2;     // 800000

  // workspace carve-out (all offsets stay 16B aligned for these sizes)
  char* ws = (char*)d_ws;
  size_t off = 0;
  float* xt = (float*)(ws + off); off += (size_t)n_nodes * CH * sizeof(float);
  float* s  = (float*)(ws + off); off += (size_t)n_nodes * HEADS * 2 * sizeof(float);
  float* eb = (float*)(ws + off); off += (size_t)n_edges * HEADS * sizeof(float);
  unsigned* mx = (unsigned*)(ws + off); off += (size_t)n_nodes * HEADS * sizeof(unsigned);
  float* sm = (float*)(ws + off);

  const int n_out = n_nodes * CH;
  const int n_nh  = n_nodes * HEADS;
  const int T = 256;

  int init_n = n_out > n_nh ? n_out : n_nh;
  gat_init<<<(init_n + T - 1) / T, T, 0, stream>>>(out, mx, sm, n_out, n_nh);

  gat_gemm_wmma<<<n_nodes / 16, 256, 0, stream>>>(x, W, xt);

  gat_node_scalars<<<(n_nh + T - 1) / T, T, 0, stream>>>(xt, a, s, n_nh);

  gat_edge_logits<<<(n_edges + T - 1) / T, T, 0, stream>>>(ei, ea, a, s, eb,
                                                           mx, n_edges);

  gat_max_fix<<<(n_nh + T - 1) / T, T, 0, stream>>>(mx, n_nh);

  gat_edge_exp<<<(n_edges + T - 1) / T, T, 0, stream>>>(
      ei, (const float*)mx, eb, sm, n_edges);

  // one wave per edge: n_edges * 32 threads
  long long agg_threads = (long long)n_edges * 32;
  gat_aggregate<<<(int)((agg_threads + T - 1) / T), T, 0, stream>>>(
      ei, xt, eb, sm, out, n_edges);
}